// Linear_8589934906
// MI455X (gfx1250) — compile-verified
//
#include <hip/hip_runtime.h>
#include <hip/hip_bf16.h>
#include <stdint.h>

// Problem constants (from the reference)
#define N_NEURONS 500000
#define P_PERIOD  16
#define F_DIM     32
#define M_SRC     200000
#define K_ROWS    10000

#define BLOCK_THREADS 128   // 4 waves (wave32)
#define NB            16    // neurons per block; 500000 / 16 = 31250 blocks exactly

typedef float v2f __attribute__((ext_vector_type(2)));
typedef float v8f __attribute__((ext_vector_type(8)));

// Low 32 bits of a generic pointer into LDS == workgroup-relative LDS byte offset.
__device__ __forceinline__ unsigned lds_off(const void* p) {
  return (unsigned)(uintptr_t)p;
}

// CDNA5 async global->LDS gathers (ASYNCcnt). Each lane copies from the global address in vaddr to the LDS address in vdst.
__device__ __forceinline__ void async_ld_b128(unsigned lds, const void* g) {
  asm volatile("global_load_async_to_lds_b128 %0, %1, off"
               :: "v"(lds), "v"(g) : "memory");
}
__device__ __forceinline__ void async_ld_b64(unsigned lds, const void* g) {
  asm volatile("global_load_async_to_lds_b64 %0, %1, off"
               :: "v"(lds), "v"(g) : "memory");
}
__device__ __forceinline__ void async_ld_b32(unsigned lds, const void* g) {
  asm volatile("global_load_async_to_lds_b32 %0, %1, off"
               :: "v"(lds), "v"(g) : "memory");
}
__device__ __forceinline__ void wait_async0() {
#if defined(__has_builtin)
#if __has_builtin(__builtin_amdgcn_s_wait_asynccnt)
  __builtin_amdgcn_s_wait_asynccnt(0);
#else
  asm volatile("s_wait_asynccnt 0" ::: "memory");
#endif
#else
  asm volatile("s_wait_asynccnt 0" ::: "memory");
#endif
}

__global__ __launch_bounds__(BLOCK_THREADS)
void gather_linear_wmma_kernel(const float* __restrict__ v0,
                               const float* __restrict__ v1,
                               const float* __restrict__ wt,
                               const int*   __restrict__ idx,
                               const int*   __restrict__ widx,
                               float*       __restrict__ out) {
  __shared__ __align__(16) int   s_idx[NB * P_PERIOD];          // [n][p]    1 KB
  __shared__ __align__(16) int   s_widx[NB];                    //           64 B
  __shared__ __align__(16) float s_w[NB * P_PERIOD];            // [n][p]    1 KB
  // Gathered values staged directly in WMMA B-fragment order:
  // [n][frag = c*4 + j*2 + h][t][col4][e]  (frag=128B, lane = t*16 + col4*4 + e)
  // where k = 4c + 2t + j (k-chunk c, VGPR parity j, lane-half t) and
  // feature f = 16h + (lane&15).
  __shared__ __align__(16) float s_val[NB * P_PERIOD * F_DIM];  // 32 KB

  const int tid  = threadIdx.x;
  const int base = blockIdx.x * NB;

  // ---- Phase 1: async-stage indices into LDS ----
  if (tid < (NB * P_PERIOD) / 4) {       // 64 threads x 16 B = 1 KB of idx
    async_ld_b128(lds_off(&s_idx[tid * 4]),
                  idx + (size_t)base * P_PERIOD + tid * 4);
  }
  if (tid < NB) {
    async_ld_b32(lds_off(&s_widx[tid]), widx + base + tid);
  }
  wait_async0();
  __syncthreads();

  // ---- Phase 2: async-gather weight rows + value rows (fragment layout) ----
  {
    // 128 threads x b64: 16 weight rows x 16 floats, layout [n][p]
    const int nl = tid >> 3, k2 = tid & 7;
    async_ld_b64(lds_off(&s_w[nl * P_PERIOD + k2 * 2]),
                 wt + (size_t)s_widx[nl] * P_PERIOD + k2 * 2);
  }
  {
    const int nl   = tid >> 3;   // neuron served by this thread (8 threads/neuron)
    const int f16  = tid & 7;    // 16-byte chunk within each 128 B row
    const int h    = f16 >> 2;   // feature half (0: f 0..15, 1: f 16..31)
    const int col4 = f16 & 3;    // float4 within the half

    int4 ridx[4];                // preload this neuron's 16 gather indices
#pragma unroll
    for (int k = 0; k < 4; ++k) ridx[k] = *(const int4*)&s_idx[nl * P_PERIOD + k * 4];
    const int* rptr = (const int*)ridx;

#pragma unroll
    for (int p = 0; p < P_PERIOD; ++p) {
      const int r = rptr[p];
      const float* g = (r < M_SRC) ? (v0 + (size_t)r * F_DIM)
                                   : (v1 + (size_t)(r - M_SRC) * F_DIM);
      const int cc   = p >> 2;         // k-chunk
      const int t    = (p >> 1) & 1;   // lane half within fragment
      const int j    = p & 1;          // B VGPR parity
      const int frag = cc * 4 + j * 2 + h;
      const unsigned dst =
          lds_off(s_val) + (unsigned)((nl * 512 + frag * 32 + t * 16 + col4 * 4) * 4);
      async_ld_b128(dst, g + f16 * 4);
    }
  }
  wait_async0();
  __syncthreads();

  // ---- Compute: per neuron, chain V_WMMA_F32_16X16X4_F32 over 4 k-chunks ----
  // A[m,k'] = w[4c + k'] for every m (broadcast)  =>  D[m,n] = y[16h+n] for all m,
  // so D[0] lane l already holds y[16h + (l&15)]: extraction = 1 cndmask + 1 store.
  const int lane = tid & 31;
  const int wave = tid >> 5;
  const int t    = lane >> 4;          // lane half -> K offset 2t in A

#pragma unroll
  for (int i = 0; i < 4; ++i) {
    const int nl = wave * 4 + i;
    v8f acc0 = {0.f, 0.f, 0.f, 0.f, 0.f, 0.f, 0.f, 0.f};
    v8f acc1 = {0.f, 0.f, 0.f, 0.f, 0.f, 0.f, 0.f, 0.f};
#pragma unroll
    for (int c = 0; c < 4; ++c) {
      // A fragment: {w[4c+2t], w[4c+2t+1]} -> one ds_read_b64 per chunk
      const v2f a = *(const v2f*)&s_w[nl * P_PERIOD + c * 4 + t * 2];
      // B fragments (128 B contiguous per register, conflict-free):
      const float* fb = &s_val[nl * 512 + c * 128 + lane];
      v2f b0, b1;
      b0.x = fb[0 * 32];   // frag(c, j=0, h=0)
      b0.y = fb[2 * 32];   // frag(c, j=1, h=0)
      b1.x = fb[1 * 32];   // frag(c, j=0, h=1)
      b1.y = fb[3 * 32];   // frag(c, j=1, h=1)
      acc0 = __builtin_amdgcn_wmma_f32_16x16x4_f32(false, a, false, b0,
                                                   (short)0, acc0, false, false);
      acc1 = __builtin_amdgcn_wmma_f32_16x16x4_f32(false, a, false, b1,
                                                   (short)0, acc1, false, false);
    }
    const float y = (lane < 16) ? acc0[0] : acc1[0];
    out[(size_t)(base + nl) * F_DIM + lane] = y;   // 128 B coalesced per wave
  }
}

extern "C" void kernel_launch(void* const* d_in, const int* in_sizes, int n_in,
                              void* d_out, int out_size, void* d_ws, size_t ws_size,
                              hipStream_t stream) {
  (void)in_sizes; (void)n_in; (void)out_size; (void)d_ws; (void)ws_size;
  const float* v0   = (const float*)d_in[0];   // values0 [M, F]
  const float* v1   = (const float*)d_in[1];   // values1 [M, F]
  const float* wt   = (const float*)d_in[2];   // w_table [K, P]
  const int*   idx  = (const int*)d_in[3];     // idx     [N, P]
  const int*   widx = (const int*)d_in[4];     // widx    [N]
  float*       out  = (float*)d_out;           // y       [N, F]

  dim3 grid(N_NEURONS / NB);                   // 31250 blocks, exact
  gather_linear_wmma_kernel<<<grid, BLOCK_THREADS, 0, stream>>>(v0, v1, wt, idx,
                                                                widx, out);
}